// SphericalFSQ_70635032150121
// MI455X (gfx1250) — compile-verified
//
#include <hip/hip_runtime.h>
#include <math.h>

typedef __attribute__((ext_vector_type(2))) float v2f;
typedef __attribute__((ext_vector_type(8))) float v8f;

#define N_CODES      512
#define N_CODE_TILES 32          // 512 codes / 16 per WMMA N-tile
#define DIM          8
#define N_RADIAL     8
#define ROW_PAD      12          // codebook row stride in LDS (dwords): conflict-free + 8B aligned

// One wave processes a 16-token tile against all 512 codes.
// A (16x4 f32): lane m (0-15) holds z_dir[token m][k,k+1] in v0,v1; lanes 16-31 hold k+2,k+3.
// B (4x16 f32): lane n (0-15) holds cb[code n][k,k+1]; lanes 16-31 hold k+2,k+3.
// C/D (16x16 f32): vgpr j, lanes 0-15 -> (token j, code lane); lanes 16-31 -> (token j+8, code lane-16).
__global__ __launch_bounds__(256)
void sfsq_wmma_kernel(const float* __restrict__ z,
                      const float* __restrict__ cb,
                      float* __restrict__ zq,
                      int* __restrict__ tok,
                      int nTiles)
{
  __shared__ float scb[N_CODES * ROW_PAD];   // 24 KB padded codebook copy

  const int tid  = threadIdx.x;
  const int lane = tid & 31;
  const int lo   = lane & 15;          // token / code index within tile
  const int hi   = lane >> 4;          // which K-half this lane supplies
  const int wavesPerBlock = blockDim.x >> 5;
  const int wave   = blockIdx.x * wavesPerBlock + (tid >> 5);
  const int nWaves = gridDim.x * wavesPerBlock;

  // ---- stage codebook into LDS with 48B row stride (bank-conflict free) ----
#pragma unroll
  for (int i = 0; i < (N_CODES * DIM) / 256; ++i) {
    const int e = i * 256 + tid;             // coalesced global read
    const int r = e >> 3, d = e & 7;
    scb[r * ROW_PAD + d] = cb[e];
  }
  __syncthreads();

  const float* myrow = scb + (size_t)lo * ROW_PAD + hi * 2;  // base for code row 'lo'

  for (int tile = wave; tile < nTiles; tile += nWaves) {
    const int tokBase = tile * 16;
    if (tile + nWaves < nTiles)
      __builtin_prefetch(z + (size_t)(tile + nWaves) * 16 * DIM, 0, 1);

    // ---- load A tile + compute magnitude / normalize ----
    const int myTok = tokBase + lo;
    const float* zrow = z + (size_t)myTok * DIM;
    v2f a0 = *(const v2f*)(zrow + hi * 2);
    v2f a1 = *(const v2f*)(zrow + 4 + hi * 2);

    float ss = a0.x * a0.x + a0.y * a0.y + a1.x * a1.x + a1.y * a1.y;
    ss += __shfl_xor(ss, 16);                 // combine the two K-halves
    const float mag = sqrtf(ss);
    const float inv = 1.0f / (mag + 1e-8f);
    a0.x *= inv; a0.y *= inv; a1.x *= inv; a1.y *= inv;

    // ---- sweep 512 codes: 32 x (2 chained f32 WMMAs) + running argmax ----
    v8f bv; int bi[8];
#pragma unroll
    for (int j = 0; j < 8; ++j) { bv[j] = -2.0f; bi[j] = 0; }

#pragma unroll 4
    for (int c = 0; c < N_CODE_TILES; ++c) {
      const float* rp = myrow + c * (16 * ROW_PAD);   // code row c*16+lo, this K-half
      const v2f b0 = *(const v2f*)(rp);               // dims {2h, 2h+1}
      const v2f b1 = *(const v2f*)(rp + 4);           // dims {4+2h, 5+2h}
      v8f acc = {};
      acc = __builtin_amdgcn_wmma_f32_16x16x4_f32(false, a0, false, b0,
                                                  (short)0, acc, false, false);
      acc = __builtin_amdgcn_wmma_f32_16x16x4_f32(false, a1, false, b1,
                                                  (short)0, acc, false, false);
      const int cand = c * 16 + lo;           // code id this lane column holds
#pragma unroll
      for (int j = 0; j < 8; ++j) {
        const bool gt = acc[j] > bv[j];       // strict > keeps first (lowest c) on ties
        bv[j] = gt ? acc[j] : bv[j];
        bi[j] = gt ? cand   : bi[j];
      }
    }

    // ---- cross-lane argmax within each 16-lane half (codes axis) ----
#pragma unroll
    for (int off = 8; off >= 1; off >>= 1) {
#pragma unroll
      for (int j = 0; j < 8; ++j) {
        const float ov = __shfl_xor(bv[j], off);
        const int   oi = __shfl_xor(bi[j], off);
        const bool take = (ov > bv[j]) || ((ov == bv[j]) && (oi < bi[j]));
        bv[j] = take ? ov : bv[j];
        bi[j] = take ? oi : bi[j];
      }
    }
    // lanes 0-15: bi[j] = argmax for token j ; lanes 16-31: token j+8

    // ---- transpose to token-major: lane t (0..15) gets argmax of token t ----
    int idx_mine = 0;
#pragma unroll
    for (int t = 0; t < 16; ++t) {
      const int srcLane = (t >> 3) << 4;      // lane 0 for tokens 0-7, lane 16 for 8-15
      const int v = __shfl(bi[t & 7], srcLane);
      idx_mine = (lane == t) ? v : idx_mine;
    }

    // ---- magnitude quantization (token-major on lanes 0-15, literal STE math) ----
    const float bounded = tanhf(mag / (0.5f * 7.0f + 1e-3f)) * 7.0f;
    float q = bounded + (rintf(bounded) - bounded);
    q = fminf(fmaxf(q, 0.0f), 7.0f);
    const int   mag_idx = (int)q;
    const float magv = 0.05f + 0.95f * (q * (1.0f / 7.0f));

    if (lane < 16) tok[myTok] = idx_mine * N_RADIAL + mag_idx;

    // ---- z_q = (z_dir + (cb[idx] - z_dir)) * mag_value, coalesced 128B stores ----
#pragma unroll
    for (int it = 0; it < 4; ++it) {
      const int e = it * 32 + lane;           // flat element in 16x8 tile
      const int t = e >> 3, d = e & 7;
      const int   ci = __shfl(idx_mine, t);
      const float mv = __shfl(magv, t);
      const float im = __shfl(inv,  t);
      const float zc   = z[(size_t)(tokBase + t) * DIM + d];
      const float zdir = zc * im;
      const float ch   = scb[(size_t)ci * ROW_PAD + d];   // gather from LDS copy
      const float dq   = zdir + (ch - zdir);  // == dir_hard up to STE rounding
      zq[(size_t)(tokBase + t) * DIM + d] = dq * mv;
    }
  }
}

extern "C" void kernel_launch(void* const* d_in, const int* in_sizes, int n_in,
                              void* d_out, int out_size, void* d_ws, size_t ws_size,
                              hipStream_t stream) {
  const float* z  = (const float*)d_in[0];        // [B, T, 8] fp32
  const float* cb = (const float*)d_in[1];        // [512, 8] fp32
  const int nZ      = in_sizes[0];                // B*T*8
  const int nTokens = nZ / DIM;                   // B*T
  const int nTiles  = nTokens / 16;               // 16 tokens per wave-tile

  float* zq = (float*)d_out;                      // output 0: z_q (fp32, nZ elems)
  int*   tk = (int*)(zq + nZ);                    // output 1: tokens (int32 bits)

  const int block = 256;                          // 8 waves per block
  const int wavesPerBlock = block / 32;
  int grid = (nTiles + wavesPerBlock * 2 - 1) / (wavesPerBlock * 2); // ~2 tiles/wave
  if (grid < 1) grid = 1;
  if (grid > 16384) grid = 16384;

  sfsq_wmma_kernel<<<grid, block, 0, stream>>>(z, cb, zq, tk, nTiles);
}